// TrueRGCNConv_9122510537206
// MI455X (gfx1250) — compile-verified
//
#include <hip/hip_runtime.h>
#include <hip/hip_bf16.h>

typedef __attribute__((ext_vector_type(2))) float v2f;
typedef __attribute__((ext_vector_type(8))) float v8f;

#define IN_DIM  128
#define OUT_DIM 128
#define N_REL   8
#define N_BASES 4
#define NKK     (IN_DIM / 4)        // 32 k-steps of K=4
#define FRAG_PER_REL (NKK * 8 * 32) // v2f elems per relation in packed B

// meta layout (ints): [0..7]=cnt, [8..15]=cursor, [16..24]=padded offsets(16-aligned)
__global__ void k_init_meta(int* meta) {
    int t = threadIdx.x;
    if (t < 32) meta[t] = 0;
}

__global__ void k_fill_perm(int* perm, int n) {
    int i = blockIdx.x * blockDim.x + threadIdx.x;
    if (i < n) perm[i] = -1;
}

// Build W[r]=sum_b coeff[r,b]*basis[b] directly in WMMA B-fragment order:
// Wp[r][kk][j][lane] = { W[r][4kk+2h][16j+n], W[r][4kk+2h+1][16j+n] },
// h=lane>>4, n=lane&15.  Loads of Bp become one coalesced b64 per fragment.
__global__ void k_build_Wpack(const float* __restrict__ coeff,
                              const float* __restrict__ basis,
                              v2f* __restrict__ Wp) {
    int tid = blockIdx.x * blockDim.x + threadIdx.x;  // < N_REL*FRAG_PER_REL exactly
    int L  = tid & 31;
    int j  = (tid >> 5) & 7;
    int kk = (tid >> 8) & 31;
    int r  = tid >> 13;
    int k  = 4 * kk + 2 * (L >> 4);
    int o  = j * 16 + (L & 15);
    float sx = 0.f, sy = 0.f;
#pragma unroll
    for (int b = 0; b < N_BASES; ++b) {
        float cb = coeff[r * N_BASES + b];
        sx += cb * basis[b * (IN_DIM * OUT_DIM) + k * OUT_DIM + o];
        sy += cb * basis[b * (IN_DIM * OUT_DIM) + (k + 1) * OUT_DIM + o];
    }
    v2f v; v.x = sx; v.y = sy;
    Wp[tid] = v;
}

// Pack w_self into the same fragment order.
__global__ void k_pack_wself(const float* __restrict__ w, v2f* __restrict__ Sp) {
    int tid = blockIdx.x * blockDim.x + threadIdx.x;  // < FRAG_PER_REL exactly
    int L  = tid & 31;
    int j  = (tid >> 5) & 7;
    int kk = (tid >> 8) & 31;
    int k  = 4 * kk + 2 * (L >> 4);
    int o  = j * 16 + (L & 15);
    v2f v; v.x = w[k * OUT_DIM + o]; v.y = w[(k + 1) * OUT_DIM + o];
    Sp[tid] = v;
}

__global__ void k_hist(const int* __restrict__ etype, int* meta, int E) {
    int i = blockIdx.x * blockDim.x + threadIdx.x;
    if (i < E) atomicAdd(&meta[etype[i]], 1);
}

// serial 8-way scan; segments aligned to 16 so each 16-edge tile is one relation
__global__ void k_scan(int* meta) {
    if (threadIdx.x == 0) {
        int off = 0;
#pragma unroll
        for (int r = 0; r < N_REL; ++r) {
            meta[16 + r] = off;
            meta[8 + r]  = off;
            off += ((meta[r] + 15) / 16) * 16;
        }
        meta[24] = off;
    }
}

__global__ void k_permute(const int* __restrict__ etype, int* meta,
                          int* __restrict__ perm, int E) {
    int i = blockIdx.x * blockDim.x + threadIdx.x;
    if (i < E) {
        int r = etype[i];
        int p = atomicAdd(&meta[8 + r], 1);
        perm[p] = i;
    }
}

// out[n,:] = x[n,:] @ w_self + bias  (fp32 WMMA, one wave = 16 rows x 128 cols)
__global__ void k_self_gemm(const float* __restrict__ x,
                            const v2f* __restrict__ Sp,
                            const float* __restrict__ bias,
                            float* __restrict__ out, int N, int nTiles) {
    int wave = (blockIdx.x * blockDim.x + threadIdx.x) >> 5;  // wave-uniform
    int lane = threadIdx.x & 31;
    if (wave >= nTiles) return;
    int m = lane & 15, half = lane >> 4, n = m;

    int row   = wave * 16 + m;
    int rowc  = row < N ? row : (N - 1);
    float msk = row < N ? 1.f : 0.f;
    const float* arow = x + (size_t)rowc * IN_DIM;

    const v8f zero = {0.f,0.f,0.f,0.f,0.f,0.f,0.f,0.f};
    v8f c[8];
#pragma unroll
    for (int j = 0; j < 8; ++j) c[j] = zero;

#pragma unroll
    for (int kk = 0; kk < NKK; ++kk) {
        int k0 = kk * 4 + 2 * half;                 // A frag: v0={K0|K2}, v1={K1|K3}
        v2f a;
        a.x = arow[k0] * msk;
        a.y = arow[k0 + 1] * msk;
#pragma unroll
        for (int j = 0; j < 8; ++j) {
            v2f b = Sp[(kk * 8 + j) * 32 + lane];   // coalesced b64
            c[j] = __builtin_amdgcn_wmma_f32_16x16x4_f32(
                       false, a, false, b, (short)0, c[j], false, false);
        }
    }
    // C/D layout: vgpr v -> M = v + 8*half, N = lane%16
#pragma unroll
    for (int v = 0; v < 8; ++v) {
        int orow = wave * 16 + v + 8 * half;
        if (orow < N) {
#pragma unroll
            for (int j = 0; j < 8; ++j) {
                int col = j * 16 + n;
                out[(size_t)orow * OUT_DIM + col] = c[j][v] + bias[col];
            }
        }
    }
}

// one wave = 16 sorted edges (single relation) x 128 outputs; atomic scatter to dst
__global__ void k_edge_gemm(const float* __restrict__ x,
                            const int* __restrict__ eidx,   // [2,E] flat
                            const v2f* __restrict__ Wp,
                            const int* __restrict__ meta,
                            const int* __restrict__ perm,
                            float* out, int E, int tileCap) {
    int wave = (blockIdx.x * blockDim.x + threadIdx.x) >> 5;
    int lane = threadIdx.x & 31;
    if (wave >= tileCap) return;
    int tb = wave * 16;
    const int* poff = meta + 16;
    if (tb >= poff[8]) return;                      // wave-uniform

    int r = 0;                                      // segment lookup (uniform)
#pragma unroll
    for (int q = 1; q < N_REL; ++q) if (tb >= poff[q]) r = q;
    const v2f* Bp = Wp + (size_t)r * FRAG_PER_REL;

    int m = lane & 15, half = lane >> 4, n = m;
    int e   = perm[tb + m];
    int ee  = e >= 0 ? e : 0;
    float msk = e >= 0 ? 1.f : 0.f;                 // sentinel rows contribute zero
    const float* arow = x + (size_t)eidx[ee] * IN_DIM;  // gather src row

    const v8f zero = {0.f,0.f,0.f,0.f,0.f,0.f,0.f,0.f};
    v8f c[8];
#pragma unroll
    for (int j = 0; j < 8; ++j) c[j] = zero;

#pragma unroll
    for (int kk = 0; kk < NKK; ++kk) {
        int k0 = kk * 4 + 2 * half;
        v2f a;
        a.x = arow[k0] * msk;
        a.y = arow[k0 + 1] * msk;
#pragma unroll
        for (int j = 0; j < 8; ++j) {
            v2f b = Bp[(kk * 8 + j) * 32 + lane];   // coalesced b64
            c[j] = __builtin_amdgcn_wmma_f32_16x16x4_f32(
                       false, a, false, b, (short)0, c[j], false, false);
        }
    }
    // scatter rows: M = v + 8*half -> atomicAdd into out[dst]
#pragma unroll
    for (int v = 0; v < 8; ++v) {
        int mm = v + 8 * half;
        int e2 = perm[tb + mm];
        if (e2 >= 0) {
            int drow = eidx[E + e2];
#pragma unroll
            for (int j = 0; j < 8; ++j)
                atomicAdd(&out[(size_t)drow * OUT_DIM + j * 16 + n], c[j][v]);
        }
    }
}

__global__ void k_relu(float* out, int n) {
    int i = blockIdx.x * blockDim.x + threadIdx.x;
    if (i < n) out[i] = fmaxf(out[i], 0.f);
}

extern "C" void kernel_launch(void* const* d_in, const int* in_sizes, int n_in,
                              void* d_out, int out_size, void* d_ws, size_t ws_size,
                              hipStream_t stream) {
    const float* x     = (const float*)d_in[0];
    const int*   eidx  = (const int*)d_in[1];   // [2, E]
    const int*   etype = (const int*)d_in[2];   // [E]
    const float* basis = (const float*)d_in[3]; // [4,128,128]
    const float* coeff = (const float*)d_in[4]; // [8,4]
    const float* wself = (const float*)d_in[5]; // [128,128]
    const float* bias  = (const float*)d_in[6]; // [128]
    float* out = (float*)d_out;

    int N = in_sizes[0] / IN_DIM;
    int E = in_sizes[2];

    v2f* Wp   = (v2f*)d_ws;                                   // 512 KB
    v2f* Sp   = Wp + (size_t)N_REL * FRAG_PER_REL;            // 64 KB
    int* meta = (int*)(Sp + FRAG_PER_REL);
    int* perm = meta + 32;
    int permCap = ((E + 15) / 16) * 16 + N_REL * 16;

    k_init_meta<<<1, 32, 0, stream>>>(meta);
    k_fill_perm<<<(permCap + 255) / 256, 256, 0, stream>>>(perm, permCap);
    k_build_Wpack<<<(N_REL * FRAG_PER_REL) / 256, 256, 0, stream>>>(coeff, basis, Wp);
    k_pack_wself<<<FRAG_PER_REL / 256, 256, 0, stream>>>(wself, Sp);
    k_hist<<<(E + 255) / 256, 256, 0, stream>>>(etype, meta, E);
    k_scan<<<1, 32, 0, stream>>>(meta);
    k_permute<<<(E + 255) / 256, 256, 0, stream>>>(etype, meta, perm, E);

    int nTiles = (N + 15) / 16;
    k_self_gemm<<<(nTiles * 32 + 127) / 128, 128, 0, stream>>>(x, Sp, bias, out, N, nTiles);

    int tileCap = permCap / 16;
    k_edge_gemm<<<(tileCap * 32 + 127) / 128, 128, 0, stream>>>(x, eidx, Wp, meta, perm,
                                                                out, E, tileCap);

    k_relu<<<(N * OUT_DIM + 255) / 256, 256, 0, stream>>>(out, N * OUT_DIM);
}